// MultiSimilarityLoss_3410204033168
// MI455X (gfx1250) — compile-verified
//
#include <hip/hip_runtime.h>
#include <math.h>

typedef __attribute__((ext_vector_type(2))) float v2f;
typedef __attribute__((ext_vector_type(8))) float v8f;

constexpr int NG   = 4;     // groups
constexpr int NB   = 4096;  // batch per group
constexpr int ND   = 256;   // feature dim
constexpr int RPB  = 16;    // rows of sim per workgroup
constexpr int NJB  = NB / 16;       // 256 column blocks of 16
constexpr int NKS  = ND / 4;        // 64 WMMA K-steps (K=4 each)
constexpr int FRAG = NKS * 64;      // 4096 floats per 16-row fragment block
constexpr int THREADS = 256;        // 8 waves
constexpr int SPAD = NB + 4;        // padded sim-strip row stride
constexpr int RSTR = ND + 4;        // padded LDS row stride in prep kernel
constexpr float MARGIN = 0.1f;
constexpr float THRESH = 0.5f;

// ---------------------------------------------------------------------------
// Kernel 1: normalize rows and emit WMMA fragment-major layout.
// For 16-row block b, K-step t (K=4), lane l (m = l&15, kh = l>>4):
//   nfF[b][t][l] = { Fn[b*16+m][4t+2kh], Fn[b*16+m][4t+2kh+1] }   (v2f)
// => every A/B fragment load in the GEMM is a coalesced 256B burst.
// ---------------------------------------------------------------------------
__global__ void ms_prep_kernel(const float* __restrict__ feats,
                               float* __restrict__ nfF,
                               float* __restrict__ out)
{
    if (blockIdx.x == 0 && threadIdx.x == 0) out[0] = 0.0f; // init accumulator

    __shared__ float rowS[RPB * RSTR];   // 16 rows x 260 floats (padded)

    const int g   = blockIdx.x >> 8;
    const int b   = blockIdx.x & 255;
    const int tid = threadIdx.x;
    const int wave = tid >> 5;
    const int lane = tid & 31;

    const float* src = feats + ((size_t)g * NB + (size_t)b * RPB) * ND;

    // phase A: coalesced load of 16x256 into padded LDS rows
    for (int idx = tid; idx < RPB * (ND / 4); idx += THREADS) {
        const int row  = idx >> 6;        // 64 float4 per row
        const int col4 = idx & 63;
        ((float4*)(rowS + row * RSTR))[col4] = ((const float4*)(src + (size_t)row * ND))[col4];
    }
    __syncthreads();

    // phase B: L2-normalize. Each half-wave (16 lanes) owns one row.
    {
        const int row = 2 * wave + (lane >> 4);
        const int c0  = (lane & 15) * 16;
        float* rp = rowS + row * RSTR + c0;
        float ss = 0.0f;
        #pragma unroll
        for (int c = 0; c < 16; ++c) ss += rp[c] * rp[c];
        #pragma unroll
        for (int off = 8; off >= 1; off >>= 1)   // reduce within 16-lane half
            ss += __shfl_xor(ss, off, 32);
        const float inv = 1.0f / fmaxf(sqrtf(ss), 1e-12f);
        #pragma unroll
        for (int c = 0; c < 16; ++c) rp[c] *= inv;
    }
    __syncthreads();

    // phase C: repack into fragment-major order, coalesced stores
    float* dst = nfF + (size_t)(g * NJB + b) * FRAG;
    for (int f = tid; f < FRAG; f += THREADS) {
        const int t  = f >> 6;
        const int l  = (f >> 1) & 31;
        const int c  = f & 1;
        const int m  = l & 15;
        const int kh = l >> 4;
        dst[f] = rowS[m * RSTR + 4 * t + 2 * kh + c];
    }
}

// ---------------------------------------------------------------------------
// Kernel 2: WMMA Gram strip in LDS + fused hard-mining passes
// ---------------------------------------------------------------------------
__global__ void ms_loss_kernel(const float* __restrict__ nfF,
                               const int*  __restrict__ labels,
                               float* __restrict__ out)
{
    extern __shared__ float smem[];
    float* simS    = smem;                       // 16 * SPAD floats (~256 KB)
    float* AfragS  = simS + RPB * SPAD;          // 4096 floats (16 KB), frag-major A
    float* redA    = AfragS + FRAG;              // 16*8
    float* redB    = redA + RPB * 8;             // 16*8
    float* minposS = redB + RPB * 8;             // 16
    float* maxnegS = minposS + RPB;              // 16
    float* posS    = maxnegS + RPB;              // 16
    float* negS    = posS + RPB;                 // 16
    int*   labI    = (int*)(negS + RPB);         // 16

    const int g    = blockIdx.x >> 8;            // 256 row-blocks per group
    const int rb   = blockIdx.x & 255;
    const int tid  = threadIdx.x;
    const int wave = tid >> 5;
    const int lane = tid & 31;

    const float* fragG = nfF + (size_t)g * NJB * FRAG;  // group's fragment blocks
    const int*   lab   = labels + g * NB;
    const int    i0    = rb * RPB;

    if (tid < RPB) labI[tid] = lab[i0 + tid];

    // ---- stage the A fragment block into LDS (one coalesced 16KB copy) ----
    {
        const float4* srcA = (const float4*)(fragG + (size_t)rb * FRAG);
        for (int idx = tid; idx < FRAG / 4; idx += THREADS)
            ((float4*)AfragS)[idx] = srcA[idx];
    }
    __syncthreads();

    // ---- GEMM: per K-step 1 ds_load_b64 (A) + 2 coalesced global_load_b64
    //      (B0/B1) + 2 v_wmma_f32_16x16x4_f32 on independent acc chains ----
    const int m  = lane & 15;
    const int kh = lane >> 4;
    const float* baseA = AfragS + lane * 2;
    for (int jb = 2 * wave; jb < NJB; jb += 16) {   // two j-blocks per iter
        const float* B0 = fragG + (size_t)jb * FRAG + lane * 2;
        const float* B1 = B0 + FRAG;
        v8f c0 = {0.f,0.f,0.f,0.f,0.f,0.f,0.f,0.f};
        v8f c1 = {0.f,0.f,0.f,0.f,0.f,0.f,0.f,0.f};
        #pragma unroll
        for (int t = 0; t < NKS; ++t) {
            v2f a  = *(const v2f*)(baseA + t * 64);  // ds_load, conflict-free
            v2f b0 = *(const v2f*)(B0 + t * 64);     // coalesced 256B, L2-hot
            v2f b1 = *(const v2f*)(B1 + t * 64);
            c0 = __builtin_amdgcn_wmma_f32_16x16x4_f32(
                     false, a, false, b0, (short)0, c0, false, false);
            c1 = __builtin_amdgcn_wmma_f32_16x16x4_f32(
                     false, a, false, b1, (short)0, c1, false, false);
        }
        const int col = jb * 16 + m;
        #pragma unroll
        for (int r = 0; r < 8; ++r) {                // C/D layout: M = r + 8*kh
            simS[(r + 8 * kh) * SPAD + col]      = c0[r];
            simS[(r + 8 * kh) * SPAD + col + 16] = c1[r];
        }
    }
    __syncthreads();

    int labR[RPB];
    #pragma unroll
    for (int r = 0; r < RPB; ++r) labR[r] = labI[r];

    // ---- pass 1: min positive / max negative per row ----
    float minp[RPB], maxn[RPB];
    #pragma unroll
    for (int r = 0; r < RPB; ++r) { minp[r] = __builtin_inff(); maxn[r] = -__builtin_inff(); }

    for (int j = tid; j < NB; j += THREADS) {
        const int lj = lab[j];
        #pragma unroll
        for (int r = 0; r < RPB; ++r) {
            const float s = simS[r * SPAD + j];
            if (lj == labR[r]) {
                if (s < 1.0f - 1e-12f) minp[r] = fminf(minp[r], s);
            } else {
                maxn[r] = fmaxf(maxn[r], s);
            }
        }
    }
    #pragma unroll
    for (int r = 0; r < RPB; ++r) {
        float v0 = minp[r], v1 = maxn[r];
        #pragma unroll
        for (int off = 16; off >= 1; off >>= 1) {
            v0 = fminf(v0, __shfl_xor(v0, off, 32));
            v1 = fmaxf(v1, __shfl_xor(v1, off, 32));
        }
        if (lane == 0) { redA[r * 8 + wave] = v0; redB[r * 8 + wave] = v1; }
    }
    __syncthreads();
    if (tid < RPB) {
        float v0 = redA[tid * 8], v1 = redB[tid * 8];
        #pragma unroll
        for (int w = 1; w < 8; ++w) {
            v0 = fminf(v0, redA[tid * 8 + w]);
            v1 = fmaxf(v1, redB[tid * 8 + w]);
        }
        minposS[tid] = v0; maxnegS[tid] = v1;
    }
    __syncthreads();

    float mpR[RPB], mnR[RPB];
    #pragma unroll
    for (int r = 0; r < RPB; ++r) { mpR[r] = minposS[r]; mnR[r] = maxnegS[r]; }

    // ---- pass 2: hard-mined exp sums ----
    float psum[RPB], nsum[RPB];
    #pragma unroll
    for (int r = 0; r < RPB; ++r) { psum[r] = 0.0f; nsum[r] = 0.0f; }

    for (int j = tid; j < NB; j += THREADS) {
        const int lj = lab[j];
        #pragma unroll
        for (int r = 0; r < RPB; ++r) {
            const float s = simS[r * SPAD + j];
            if (lj != labR[r]) {
                if (s + MARGIN > mpR[r]) nsum[r] += expf(40.0f * (s - THRESH));
            } else if (s < 1.0f - 1e-12f) {
                if (s - MARGIN < mnR[r]) psum[r] += expf(-2.0f * (s - THRESH));
            }
        }
    }
    #pragma unroll
    for (int r = 0; r < RPB; ++r) {
        float v0 = psum[r], v1 = nsum[r];
        #pragma unroll
        for (int off = 16; off >= 1; off >>= 1) {
            v0 += __shfl_xor(v0, off, 32);
            v1 += __shfl_xor(v1, off, 32);
        }
        if (lane == 0) { redA[r * 8 + wave] = v0; redB[r * 8 + wave] = v1; }
    }
    __syncthreads();
    if (tid < RPB) {
        float ps = 0.0f, ns = 0.0f;
        #pragma unroll
        for (int w = 0; w < 8; ++w) { ps += redA[tid * 8 + w]; ns += redB[tid * 8 + w]; }
        posS[tid] = ps; negS[tid] = ns;
    }
    __syncthreads();
    if (tid < RPB) {
        const bool valid1 = (minposS[tid] < __builtin_inff()) &&
                            (maxnegS[tid] > -__builtin_inff());
        const float ps = posS[tid], ns = negS[tid];
        const bool valid2 = (ps > 0.0f) && (ns > 0.0f);  // exp > 0 => any-kept
        const float rl = (valid1 && valid2)
                           ? (0.5f * log1pf(ps) + 0.025f * log1pf(ns))
                           : 0.0f;
        redA[tid] = rl;   // reuse as loss scratch
    }
    __syncthreads();
    if (tid == 0) {
        float t = 0.0f;
        #pragma unroll
        for (int r = 0; r < RPB; ++r) t += redA[r];
        atomicAdd(out, t * (1.0f / (float)(NG * NB)));   // mean over groups of sum/B
    }
}

extern "C" void kernel_launch(void* const* d_in, const int* in_sizes, int n_in,
                              void* d_out, int out_size, void* d_ws, size_t ws_size,
                              hipStream_t stream) {
    const float* feats  = (const float*)d_in[0];   // [4,4096,256] f32
    const int*   labels = (const int*)d_in[1];     // [4,4096] int32 (JAX x64 off)
    float* out = (float*)d_out;
    float* nfF = (float*)d_ws;                     // fragment-major normalized feats: 16.8 MB

    // 1) normalize + repack to WMMA fragment-major layout; zero the accumulator
    ms_prep_kernel<<<NG * NJB, THREADS, 0, stream>>>(feats, nfF, out);

    // 2) fused WMMA Gram-strip (LDS-resident) + mining passes
    const size_t shmem =
        (size_t)(RPB * SPAD + FRAG + RPB * 8 * 2 + RPB * 4 + RPB) * sizeof(float);
    ms_loss_kernel<<<NG * NJB, THREADS, shmem, stream>>>(nfF, labels, out);
}